// ConvNext_82617990906073
// MI455X (gfx1250) — compile-verified
//
#include <hip/hip_runtime.h>
#include <hip/hip_bf16.h>

typedef __attribute__((ext_vector_type(2))) float v2f;
typedef __attribute__((ext_vector_type(8))) float v8f;

#define F_DIM 128
#define A_DIM 64
#define HID   512
#define LN_EPS 1e-5f

__device__ __forceinline__ v8f wmma_f32(v2f a, v2f b, v8f c) {
    // D = A(16x4 f32) * B(4x16 f32) + C(16x16 f32)
    return __builtin_amdgcn_wmma_f32_16x16x4_f32(
        false, a, false, b, (short)0, c, false, false);
}

__device__ __forceinline__ float gelu_exact(float v) {
    return 0.5f * v * (1.0f + erff(v * 0.70710678118654752f));
}

// ---------------------------------------------------------------------------
// Kernel A: edge-conditioned conv.  kernel = attr @ W  (WMMA f32),
// msgs = x[src]*kernel, scatter-add into h[tgt] with f32 global atomics.
// One wave == 16 edges x 128 features.
// kernel_W staged TRANSPOSED into LDS via global_load_async_to_lds_b32
// (ASYNCcnt): (k,k+1) contiguous -> move-free ds_load_b64 B operands.
// ---------------------------------------------------------------------------
__global__ __launch_bounds__(256) void edge_conv_kernel(
    const float* __restrict__ x, const float* __restrict__ attr,
    const long long* __restrict__ edge_index, const float* __restrict__ kernel_W,
    float* __restrict__ h, int nEdges, int nTiles)
{
    // sW[n][k], row stride 68 floats: bank advance 4 per n -> half-0 lanes
    // (k%4 in {0,1}) and half-1 lanes (k%4 in {2,3}) hit disjoint bank sets,
    // conflict-free b64 reads; stride even keeps 8B alignment for even k.
    __shared__ __align__(16) float sW[F_DIM * 68];
    for (int i = threadIdx.x; i < A_DIM * F_DIM; i += blockDim.x) {
        int k = i >> 7;          // kernel_W row (0..63)
        int n = i & (F_DIM - 1); // kernel_W col (0..127)
        unsigned lds_off = (unsigned)(size_t)&sW[n * 68 + k];
        unsigned long long ga = (unsigned long long)(size_t)(kernel_W + i);
        asm volatile("global_load_async_to_lds_b32 %0, %1, off"
                     :: "v"(lds_off), "v"(ga) : "memory");
    }
    asm volatile("s_wait_asynccnt 0" ::: "memory");
    __syncthreads();

    int wave = blockIdx.x * (blockDim.x >> 5) + (threadIdx.x >> 5);
    if (wave >= nTiles) return;
    int lane = threadIdx.x & 31;
    int lrow = lane & 15;   // A-matrix row / B,C column-within-tile
    int half = lane >> 4;   // K phase select

    long long e0 = (long long)wave * 16;

    // Load scatter/gather indices up front and prefetch the x[src] rows the
    // epilogue will gather: lanes stride 32B so 16 lanes cover all 8 lines
    // of each 512B feature row.  The 128-WMMA loop below hides the latency.
    const long long* srcIdx = edge_index;
    const long long* tgtIdx = edge_index + nEdges;
    long long sIdx[8], tIdx[8];
    #pragma unroll
    for (int v = 0; v < 8; ++v) {
        long long e = e0 + v + half * 8;
        sIdx[v] = srcIdx[e];
        tIdx[v] = tgtIdx[e];
        __builtin_prefetch(x + sIdx[v] * F_DIM + lrow * 8, 0, 0);
    }

    const float* arow = attr + (e0 + lrow) * A_DIM + half * 2;
    v8f acc[8] = {};
    #pragma unroll
    for (int kb = 0; kb < A_DIM / 4; ++kb) {           // 16 K-steps of 4
        v2f a;
        a.x = arow[kb * 4 + 0];
        a.y = arow[kb * 4 + 1];
        #pragma unroll
        for (int nb = 0; nb < 8; ++nb) {               // 8 N-tiles of 16
            const float* bp = &sW[(nb * 16 + lrow) * 68 + kb * 4 + half * 2];
            v2f b; b.x = bp[0]; b.y = bp[1];
            acc[nb] = wmma_f32(a, b, acc[nb]);
        }
    }

    #pragma unroll
    for (int v = 0; v < 8; ++v) {
        const float* xrow = x + sIdx[v] * F_DIM + lrow;
        float*       hrow = h + tIdx[v] * F_DIM + lrow;
        #pragma unroll
        for (int nb = 0; nb < 8; ++nb) {
            float msg = xrow[nb * 16] * acc[nb][v];
            __hip_atomic_fetch_add(&hrow[nb * 16], msg,
                                   __ATOMIC_RELAXED, __HIP_MEMORY_SCOPE_AGENT);
        }
    }
}

// ---------------------------------------------------------------------------
// Kernel B: h = LN(h + conv_bias) in place.  One wave per node, wave32
// shuffle-tree reductions (4 features per lane).
// ---------------------------------------------------------------------------
__global__ __launch_bounds__(256) void ln_kernel(
    float* __restrict__ h, const float* __restrict__ conv_bias,
    const float* __restrict__ gamma, const float* __restrict__ beta, int nNodes)
{
    int node = blockIdx.x * 8 + (threadIdx.x >> 5);
    if (node >= nNodes) return;
    int lane = threadIdx.x & 31;
    float* row = h + (long long)node * F_DIM + lane * 4;

    float4 hv = *(const float4*)row;
    float4 bv = *(const float4*)(conv_bias + lane * 4);
    hv.x += bv.x; hv.y += bv.y; hv.z += bv.z; hv.w += bv.w;

    float s = hv.x + hv.y + hv.z + hv.w;
    #pragma unroll
    for (int m = 16; m >= 1; m >>= 1) s += __shfl_xor(s, m, 32);
    float mu = s * (1.0f / F_DIM);

    float dx = hv.x - mu, dy = hv.y - mu, dz = hv.z - mu, dw = hv.w - mu;
    float q = dx * dx + dy * dy + dz * dz + dw * dw;
    #pragma unroll
    for (int m = 16; m >= 1; m >>= 1) q += __shfl_xor(q, m, 32);
    float rinv = rsqrtf(q * (1.0f / F_DIM) + LN_EPS);

    float4 g = *(const float4*)(gamma + lane * 4);
    float4 b = *(const float4*)(beta + lane * 4);
    float4 o;
    o.x = dx * rinv * g.x + b.x;
    o.y = dy * rinv * g.y + b.y;
    o.z = dz * rinv * g.z + b.z;
    o.w = dw * rinv * g.w + b.w;
    *(float4*)row = o;
}

// ---------------------------------------------------------------------------
// Kernel C: out = x + layer_scale * (GELU(h@W1 + b1) @ W2 + b2)
// One wave per 16-node strip; both GEMMs on v_wmma_f32_16x16x4_f32.
// Intermediate C->A relayout goes through a small per-wave LDS tile.
// ---------------------------------------------------------------------------
#define WAVES_C 4
__global__ __launch_bounds__(32 * WAVES_C) void mlp_kernel(
    const float* __restrict__ x, const float* __restrict__ hln,
    const float* __restrict__ W1, const float* __restrict__ b1,
    const float* __restrict__ W2, const float* __restrict__ b2,
    const float* __restrict__ layer_scale, float* __restrict__ out, int nTiles)
{
    __shared__ __align__(16) float shA[WAVES_C][16 * 132]; // h tile, pad 132
    __shared__ __align__(16) float shS[WAVES_C][16 * 18];  // GELU staging, pad 18

    int wslot = threadIdx.x >> 5;
    int wave  = blockIdx.x * WAVES_C + wslot;
    if (wave >= nTiles) return;        // per-wave LDS regions, no block barrier
    int lane = threadIdx.x & 31;
    int lrow = lane & 15;
    int half = lane >> 4;

    long long n0 = (long long)wave * 16;
    float* A = shA[wslot];
    float* S = shS[wslot];

    // Stage 16x128 h_ln tile (float4 in / float4 out, stride 132 stays aligned)
    for (int i = lane; i < 16 * 32; i += 32) {
        int r  = i >> 5;
        int c4 = (i & 31) * 4;
        float4 hv = *(const float4*)(hln + (n0 + r) * F_DIM + c4);
        *(float4*)&A[r * 132 + c4] = hv;
    }
    asm volatile("s_wait_dscnt 0" ::: "memory");

    v8f accO[8] = {};
    for (int nt = 0; nt < HID / 16; ++nt) {            // 32 intermediate tiles
        v8f acc1 = {};
        #pragma unroll
        for (int kb = 0; kb < F_DIM / 4; ++kb) {       // GEMM1: K=128
            int k = kb * 4 + half * 2;
            v2f a; a.x = A[lrow * 132 + k]; a.y = A[lrow * 132 + k + 1];
            v2f b;
            b.x = W1[(k + 0) * HID + nt * 16 + lrow];
            b.y = W1[(k + 1) * HID + nt * 16 + lrow];
            acc1 = wmma_f32(a, b, acc1);
        }
        // bias + exact GELU, park in LDS for C->A relayout
        float bias = b1[nt * 16 + lrow];
        #pragma unroll
        for (int v = 0; v < 8; ++v) {
            float val = gelu_exact(acc1[v] + bias);
            S[(v + half * 8) * 18 + lrow] = val;
        }
        asm volatile("s_wait_dscnt 0" ::: "memory");   // cross-lane LDS dep

        #pragma unroll
        for (int j = 0; j < 4; ++j) {                  // GEMM2: this K block
            int sk = j * 4 + half * 2;
            int kk = nt * 16 + sk;
            v2f a2;
            a2.x = S[lrow * 18 + sk];
            a2.y = S[lrow * 18 + sk + 1];
            #pragma unroll
            for (int nb = 0; nb < 8; ++nb) {
                v2f b2v;
                b2v.x = W2[(kk + 0) * F_DIM + nb * 16 + lrow];
                b2v.y = W2[(kk + 1) * F_DIM + nb * 16 + lrow];
                accO[nb] = wmma_f32(a2, b2v, accO[nb]);
            }
        }
        asm volatile("s_wait_dscnt 0" ::: "memory");   // before S is rewritten
    }

    // Epilogue: + b2, * layer_scale, + residual x
    #pragma unroll
    for (int v = 0; v < 8; ++v) {
        long long node = n0 + v + half * 8;
        #pragma unroll
        for (int nb = 0; nb < 8; ++nb) {
            int col = nb * 16 + lrow;
            float val = accO[nb][v] + b2[col];
            val = layer_scale[col] * val + x[node * F_DIM + col];
            out[node * F_DIM + col] = val;
        }
    }
}

// ---------------------------------------------------------------------------
extern "C" void kernel_launch(void* const* d_in, const int* in_sizes, int n_in,
                              void* d_out, int out_size, void* d_ws, size_t ws_size,
                              hipStream_t stream) {
    const float*     x          = (const float*)d_in[0];
    const float*     attr       = (const float*)d_in[1];
    const long long* edge_index = (const long long*)d_in[2];
    /* d_in[3] = batch (unused: single graph) */
    const float*     kernel_W   = (const float*)d_in[4];
    const float*     conv_bias  = (const float*)d_in[5];
    const float*     ln_gamma   = (const float*)d_in[6];
    const float*     ln_beta    = (const float*)d_in[7];
    const float*     W1         = (const float*)d_in[8];
    const float*     b1         = (const float*)d_in[9];
    const float*     W2         = (const float*)d_in[10];
    const float*     b2         = (const float*)d_in[11];
    const float*     lscale     = (const float*)d_in[12];
    float*           outp       = (float*)d_out;

    int nNodes = in_sizes[0] / F_DIM;   // 50000
    int nEdges = in_sizes[1] / A_DIM;   // 600000

    float* h = (float*)d_ws;            // [nNodes, 128] accumulator / LN buffer
    hipMemsetAsync(h, 0, (size_t)nNodes * F_DIM * sizeof(float), stream);

    int tilesA = (nEdges + 15) / 16;
    edge_conv_kernel<<<(tilesA + 7) / 8, 256, 0, stream>>>(
        x, attr, edge_index, kernel_W, h, nEdges, tilesA);

    ln_kernel<<<(nNodes + 7) / 8, 256, 0, stream>>>(
        h, conv_bias, ln_gamma, ln_beta, nNodes);

    int tilesC = (nNodes + 15) / 16;
    mlp_kernel<<<(tilesC + WAVES_C - 1) / WAVES_C, 32 * WAVES_C, 0, stream>>>(
        x, h, W1, b1, W2, b2, lscale, outp, tilesC);
}